// NCE_36163624632492
// MI455X (gfx1250) — compile-verified
//
#include <hip/hip_runtime.h>
#include <hip/hip_bf16.h>
#include <stdint.h>

#define N_PTS 2048
#define D_DIM 64
#define TILE_J 128                                 // x rows staged per LDS tile
#define ROW_STRIDE 68                              // floats; 272B: 16B-aligned + conflict-free b128
#define BLK 256                                    // 8 waves (wave32)
#define ROWS_PER_BLK 8                             // one y-row per wave
#define NTILES (N_PTS / TILE_J)                    // 16
#define ISSUES_PER_THREAD (TILE_J * 16 / BLK)      // 8 async b128 per thread per tile
#define LDS_BYTES (2u * TILE_J * ROW_STRIDE * 4u)  // double buffered

#define GLOBAL_AS __attribute__((address_space(1)))
#define LDS_AS    __attribute__((address_space(3)))

typedef int v4i __attribute__((vector_size(16)));

__device__ __forceinline__ void async_copy_b128(const void* gsrc, void* ldst) {
#if __has_builtin(__builtin_amdgcn_global_load_async_to_lds_b128)
  __builtin_amdgcn_global_load_async_to_lds_b128(
      (GLOBAL_AS v4i*)gsrc, (LDS_AS v4i*)ldst, 0, 0);
#else
  uint32_t l32 = (uint32_t)(uintptr_t)ldst;
  asm volatile("global_load_async_to_lds_b128 %0, %1, off"
               :: "v"(l32), "v"(gsrc) : "memory");
#endif
}

template <int NW>
__device__ __forceinline__ void wait_asynccnt() {
#if __has_builtin(__builtin_amdgcn_s_wait_asynccnt)
  __builtin_amdgcn_s_wait_asynccnt(NW);
#else
  asm volatile("s_wait_asynccnt %0" :: "i"(NW));
#endif
}

__device__ __forceinline__ float fast_exp2(float x) {
#if __has_builtin(__builtin_amdgcn_exp2f)
  return __builtin_amdgcn_exp2f(x);   // v_exp_f32 (native base-2)
#else
  return exp2f(x);
#endif
}

__device__ __forceinline__ float fast_log2(float x) {
#if __has_builtin(__builtin_amdgcn_logf)
  return __builtin_amdgcn_logf(x);    // v_log_f32 (native log2)
#else
  return __log2f(x);
#endif
}

__global__ __launch_bounds__(BLK) void nce_rows_kernel(
    const float* __restrict__ x, const float* __restrict__ y,
    float* __restrict__ row_out) {
  extern __shared__ float lds[];
  const int tid  = threadIdx.x;
  const int lane = tid & 31;
  const int wave = tid >> 5;
  // one row per wave; force wave-uniformity so y loads scalarize (s_load)
  const int i = __builtin_amdgcn_readfirstlane((int)blockIdx.x * ROWS_PER_BLK + wave);

  // preload y row (64 floats, wave-uniform)
  float4 yv[16];
  {
    const float4* y4 = (const float4*)(y + i * D_DIM);
#pragma unroll
    for (int t = 0; t < 16; ++t) yv[t] = y4[t];
  }

  // async-stage one x tile: 128 rows x 64 f32, padded to 68-float LDS rows
  auto issue_tile = [&](int tile, int buf) {
    float* base = lds + buf * (TILE_J * ROW_STRIDE);
    const float* gx = x + (size_t)tile * TILE_J * D_DIM;
#pragma unroll
    for (int k = 0; k < ISSUES_PER_THREAD; ++k) {
      int c = tid + k * BLK;      // 0..2047
      int row = c >> 4;           // 0..127
      int seg = c & 15;           // 16B segment within 256B row
      async_copy_b128(gx + row * D_DIM + seg * 4,
                      base + row * ROW_STRIDE + seg * 4);
    }
  };

  float m = -3.0e30f, s = 0.0f, diag = 0.0f;  // log2-domain flash-LSE state

  issue_tile(0, 0);
  for (int t = 0; t < NTILES; ++t) {
    if (t + 1 < NTILES) {
      issue_tile(t + 1, (t + 1) & 1);
      wait_asynccnt<ISSUES_PER_THREAD>();  // in-order: tile t fully landed
    } else {
      wait_asynccnt<0>();
    }
    __syncthreads();
    const float* xb = lds + (t & 1) * (TILE_J * ROW_STRIDE);
#pragma unroll 1
    for (int jt = 0; jt < TILE_J / 32; ++jt) {
      const int jl = jt * 32 + lane;
      const int j  = t * TILE_J + jl;
      const float4* xr = (const float4*)(xb + jl * ROW_STRIDE);
      // S2 = sum_d log2(1 + z^2), via products of 8 terms (fp32-safe)
      float S2 = 0.0f;
#pragma unroll
      for (int g = 0; g < 8; ++g) {
        float4 a = xr[2 * g];
        float4 b = xr[2 * g + 1];
        float4 ya = yv[2 * g];
        float4 yb = yv[2 * g + 1];
        float z, p;
        z = ya.x - a.x; p  = fmaf(z, z, 1.0f);
        z = ya.y - a.y; p *= fmaf(z, z, 1.0f);
        z = ya.z - a.z; p *= fmaf(z, z, 1.0f);
        z = ya.w - a.w; p *= fmaf(z, z, 1.0f);
        z = yb.x - b.x; p *= fmaf(z, z, 1.0f);
        z = yb.y - b.y; p *= fmaf(z, z, 1.0f);
        z = yb.z - b.z; p *= fmaf(z, z, 1.0f);
        z = yb.w - b.w; p *= fmaf(z, z, 1.0f);
        S2 += fast_log2(p);
      }
      float v = -S2;
      if (j == i) { diag = S2; v = -3.0e30f; }  // exclude diagonal from LSE
      float mn = fmaxf(m, v);
      s = s * fast_exp2(m - mn) + fast_exp2(v - mn);
      m = mn;
    }
    __syncthreads();
  }

  // wave32 butterfly combine of (m,s) pairs + diagonal
#pragma unroll
  for (int off = 16; off >= 1; off >>= 1) {
    float m2 = __shfl_xor(m, off, 32);
    float s2 = __shfl_xor(s, off, 32);
    float d2 = __shfl_xor(diag, off, 32);
    float mn = fmaxf(m, m2);
    s = s * fast_exp2(m - mn) + s2 * fast_exp2(m2 - mn);
    m = mn;
    diag += d2;
  }
  if (lane == 0) row_out[i] = diag + m + fast_log2(s);  // log2 units
}

__global__ __launch_bounds__(256) void nce_reduce_kernel(
    const float* __restrict__ row_vals, float* __restrict__ out) {
  __shared__ float wsum[8];
  const int tid = threadIdx.x;
  float a = 0.0f;
#pragma unroll
  for (int k = 0; k < N_PTS / 256; ++k) a += row_vals[tid + k * 256];
#pragma unroll
  for (int off = 16; off >= 1; off >>= 1) a += __shfl_xor(a, off, 32);
  if ((tid & 31) == 0) wsum[tid >> 5] = a;
  __syncthreads();
  if (tid == 0) {
    float tot = 0.0f;
#pragma unroll
    for (int w = 0; w < 8; ++w) tot += wsum[w];
    const float LN2 = 0.69314718055994530942f;
    out[0] = tot * (LN2 / (float)N_PTS) - logf((float)(N_PTS - 1));
  }
}

extern "C" void kernel_launch(void* const* d_in, const int* in_sizes, int n_in,
                              void* d_out, int out_size, void* d_ws, size_t ws_size,
                              hipStream_t stream) {
  const float* x = (const float*)d_in[0];  // [2048, 64] f32
  const float* y = (const float*)d_in[1];  // [2048, 64] f32
  float* row_vals = (float*)d_ws;          // 2048 f32 scratch
  float* out = (float*)d_out;              // 1 f32

  nce_rows_kernel<<<N_PTS / ROWS_PER_BLK, BLK, LDS_BYTES, stream>>>(x, y, row_vals);
  nce_reduce_kernel<<<1, 256, 0, stream>>>(row_vals, out);
}